// SSIMLoss_82343112999193
// MI455X (gfx1250) — compile-verified
//
#include <hip/hip_runtime.h>

// ---------------------------------------------------------------------------
// SSIM loss, MI455X (gfx1250). Separable 11-tap Gaussian conv expressed as
// banded matmuls on V_WMMA_F32_16X16X4_F32 (fp32 matrix path, keeps reference
// precision). Memory floor ~4.3us (100MB @ 23.3TB/s); both images fit in the
// 192MB L2 so halo re-reads are cheap.
// ---------------------------------------------------------------------------

typedef __attribute__((ext_vector_type(2))) float v2f;
typedef __attribute__((ext_vector_type(8))) float v8f;

#define IMG        512
#define TS         16          // output tile edge
#define PADW       5           // (11-1)/2
#define KROWS      26          // TS + 10 halo rows
#define LDS_ROWS   32          // padded to 2 x 16 row-groups
#define LDS_COLS   28          // 26 halo cols padded to 7 K-chunks of 4
#define NKCHUNK    7           // ceil(28/4)
#define TILES      32          // 512/16 per dimension
#define PLANES     48          // 16 batch * 3 channels
#define NWAVES     (PLANES * TILES * TILES)      // 49152
#define WPB        4           // waves per block
#define NBLOCKS    (NWAVES / WPB)                // 12288
#define NTOTAL     (16.0f * 3.0f * 512.0f * 512.0f)

// gaussian over arange(11) (NOT centered — faithful to reference), normalized
__constant__ float GW[11] = {
    0.4201733f,    0.3364490f,    0.1727390f,    0.0568650f,
    0.0120023f,    1.624370e-3f,  1.409530e-4f,  7.842290e-6f,
    2.797504e-7f,  6.398975e-9f,  9.216630e-11f
};

__device__ __forceinline__ float bandw(int k, int x) {
    int d = k - x;
    return (d >= 0 && d <= 10) ? GW[d] : 0.0f;
}

__device__ __forceinline__ float combine_q(int q, float a, float b) {
    switch (q) {
        case 0:  return a;
        case 1:  return b;
        case 2:  return a * a;
        case 3:  return b * b;
        default: return a * b;
    }
}

__global__ __launch_bounds__(WPB * 32, 1)
void ssim_tiles(const float* __restrict__ img1, const float* __restrict__ img2,
                float* __restrict__ partial) {
    // per-wave private LDS slice: in1[32][28] + in2[32][28] + R[32][16]
    __shared__ float lds[WPB][2 * LDS_ROWS * LDS_COLS + LDS_ROWS * TS];

    const int lane = threadIdx.x & 31;
    const int wv   = threadIdx.x >> 5;
    const int half = lane >> 4;       // 0: lanes 0-15, 1: lanes 16-31
    const int nidx = lane & 15;

    float* in1 = &lds[wv][0];
    float* in2 = in1 + LDS_ROWS * LDS_COLS;
    float* R   = in2 + LDS_ROWS * LDS_COLS;

    const int waveId = blockIdx.x * WPB + wv;
    const int plane  = waveId / (TILES * TILES);
    const int tid    = waveId % (TILES * TILES);
    const int y0     = (tid / TILES) * TS;
    const int x0     = (tid % TILES) * TS;

    const float* p1 = img1 + (size_t)plane * IMG * IMG;
    const float* p2 = img2 + (size_t)plane * IMG * IMG;

    // ---- load halo tile (zero-padded at image borders; OOB cols/rows of the
    // padded LDS tile are killed by zero band weights but must stay finite) ----
    for (int i = lane; i < LDS_ROWS * LDS_COLS; i += 32) {   // 896/32: uniform
        int r  = i / LDS_COLS;
        int k  = i % LDS_COLS;
        int gy = y0 + r - PADW;
        int gx = x0 + k - PADW;
        bool ok = (gy >= 0) & (gy < IMG) & (gx >= 0) & (gx < IMG);
        size_t off = (size_t)gy * IMG + gx;
        in1[i] = ok ? p1[off] : 0.0f;
        in2[i] = ok ? p2[off] : 0.0f;
    }

    // ---- band weights: W[k][x] = g[k-x]. Identical per-lane values serve as
    // the B operand of the row pass and the A operand of the column pass. ----
    v2f wband[NKCHUNK];
#pragma unroll
    for (int kk = 0; kk < NKCHUNK; ++kk) {
        int kb = 4 * kk + 2 * half;
        wband[kk].x = bandw(kb,     nidx);
        wband[kk].y = bandw(kb + 1, nidx);
    }

    v8f res[5];

#pragma unroll
    for (int q = 0; q < 5; ++q) {
        // ---- row pass: R[m][x] = sum_k S[m][k] * g[k-x], 26 rows as 2 groups
#pragma unroll
        for (int g = 0; g < 2; ++g) {
            const int r0 = g * 16;
            v8f acc = {0.f, 0.f, 0.f, 0.f, 0.f, 0.f, 0.f, 0.f};
#pragma unroll
            for (int kk = 0; kk < NKCHUNK; ++kk) {
                int kb  = 4 * kk + 2 * half;
                int row = (r0 + nidx) * LDS_COLS;
                v2f a;
                a.x = combine_q(q, in1[row + kb],     in2[row + kb]);
                a.y = combine_q(q, in1[row + kb + 1], in2[row + kb + 1]);
                acc = __builtin_amdgcn_wmma_f32_16x16x4_f32(
                    false, a, false, wband[kk], (short)0, acc, false, false);
            }
            // D layout: VGPR v, lane -> (M = v + 8*half, N = nidx)
#pragma unroll
            for (int v = 0; v < 8; ++v)
                R[(r0 + v + 8 * half) * TS + nidx] = acc[v];
        }
        // same-wave DS ops complete in order: no barrier needed intra-wave

        // ---- column pass: Out[y][x] = sum_k g[k-y] * R[k][x] ----
        {
            v8f acc = {0.f, 0.f, 0.f, 0.f, 0.f, 0.f, 0.f, 0.f};
#pragma unroll
            for (int kk = 0; kk < NKCHUNK; ++kk) {
                int kb = 4 * kk + 2 * half;
                v2f b;
                b.x = R[kb * TS + nidx];
                b.y = R[(kb + 1) * TS + nidx];
                acc = __builtin_amdgcn_wmma_f32_16x16x4_f32(
                    false, wband[kk], false, b, (short)0, acc, false, false);
            }
            res[q] = acc;
        }
    }

    // ---- elementwise SSIM on the five accumulators (shared C/D layout) ----
    const float C1 = 1.0e-4f, C2 = 9.0e-4f;
    float lsum = 0.0f;
#pragma unroll
    for (int e = 0; e < 8; ++e) {
        float mu1 = res[0][e], mu2 = res[1][e];
        float s11 = res[2][e] - mu1 * mu1;
        float s22 = res[3][e] - mu2 * mu2;
        float s12 = res[4][e] - mu1 * mu2;
        float num = (2.0f * mu1 * mu2 + C1) * (2.0f * s12 + C2);
        float den = (mu1 * mu1 + mu2 * mu2 + C1) * (s11 + s22 + C2);
        lsum += num / den;
    }

    // wave32 reduction
#pragma unroll
    for (int m = 16; m >= 1; m >>= 1)
        lsum += __shfl_xor(lsum, m, 32);

    __shared__ float wsum[WPB];
    if (lane == 0) wsum[wv] = lsum;
    __syncthreads();
    if (threadIdx.x == 0) {
        float s = 0.0f;
#pragma unroll
        for (int w = 0; w < WPB; ++w) s += wsum[w];
        partial[blockIdx.x] = s;
    }
}

__global__ __launch_bounds__(256, 1)
void ssim_final(const float* __restrict__ partial, int n, float* __restrict__ out) {
    __shared__ float s[256];
    float acc = 0.0f;
    for (int i = threadIdx.x; i < n; i += 256) acc += partial[i];
    s[threadIdx.x] = acc;
    __syncthreads();
    for (int st = 128; st > 0; st >>= 1) {
        if ((int)threadIdx.x < st) s[threadIdx.x] += s[threadIdx.x + st];
        __syncthreads();
    }
    if (threadIdx.x == 0) out[0] = 1.0f - s[0] / NTOTAL;
}

extern "C" void kernel_launch(void* const* d_in, const int* in_sizes, int n_in,
                              void* d_out, int out_size, void* d_ws, size_t ws_size,
                              hipStream_t stream) {
    (void)in_sizes; (void)n_in; (void)out_size; (void)ws_size;
    const float* img1 = (const float*)d_in[0];
    const float* img2 = (const float*)d_in[1];
    float* partial = (float*)d_ws;              // NBLOCKS floats = 48 KB
    float* out = (float*)d_out;

    ssim_tiles<<<NBLOCKS, WPB * 32, 0, stream>>>(img1, img2, partial);
    ssim_final<<<1, 256, 0, stream>>>(partial, NBLOCKS, out);
}